// AdjSMP_69329362092564
// MI455X (gfx1250) — compile-verified
//
#include <hip/hip_runtime.h>

typedef __attribute__((ext_vector_type(2))) float v2f;
typedef __attribute__((ext_vector_type(8))) float v8f;

// ---------------- degree / dinv ----------------
__global__ void init_deg_kernel(float* deg, int n) {
  int i = blockIdx.x * blockDim.x + threadIdx.x;
  if (i < n) deg[i] = 1.0f;  // self loop contributes 1
}

__global__ void edge_deg_kernel(const long long* __restrict__ row, float* deg, int e) {
  int i = blockIdx.x * blockDim.x + threadIdx.x;
  if (i < e) atomicAdd(&deg[(int)row[i]], 1.0f);
}

__global__ void dinv_kernel(float* deg, int n) {
  int i = blockIdx.x * blockDim.x + threadIdx.x;
  if (i < n) deg[i] = 1.0f / sqrtf(deg[i]);
}

// ---------------- GEMM1: h0 = x @ W_sgc + b  (N x 256 x 128) ----------------
// block = 256 threads = 8 waves; each block does a 16x128 output tile.
// A tile (16x256) staged in LDS via CDNA5 async copy (ASYNCcnt path);
// all 8 column-tile waves share it.
__global__ void __launch_bounds__(256)
gemm_sgc_wmma(const float* __restrict__ X, const float* __restrict__ W,
              const float* __restrict__ bias, float* __restrict__ H, int n) {
  __shared__ float As[16 * 256];  // 16 KB
  int row0 = blockIdx.x << 4;
  if (row0 >= n) return;

  // async global->LDS staging: 1024 x b128, 4 per thread, no VGPR bounce
  {
    unsigned ldsBase = (unsigned)(uintptr_t)As;  // low 32 bits = LDS offset
    const float4* src = (const float4*)(X + (size_t)row0 * 256);
    for (int t = threadIdx.x; t < 16 * 64; t += 256) {
      unsigned la = ldsBase + (unsigned)t * 16u;
      const float4* ga = src + t;
      asm volatile("global_load_async_to_lds_b128 %0, %1, off"
                   :: "v"(la), "v"(ga) : "memory");
    }
    asm volatile("s_wait_asynccnt 0x0" ::: "memory");
  }
  __syncthreads();

  int lane = threadIdx.x & 31;
  int wave = threadIdx.x >> 5;  // column tile 0..7
  int lm = lane & 15;
  int hi = lane >> 4;           // half-wave selects K+0/1 vs K+2/3
  int col = (wave << 4) + lm;

  v8f c;
  float b0 = bias[col];
#pragma unroll
  for (int r = 0; r < 8; ++r) c[r] = b0;

  const float* a_base = As + lm * 256 + (hi << 1);
  const float* w_base = W + (size_t)(hi << 1) * 128 + col;

#pragma unroll 4
  for (int k = 0; k < 256; k += 4) {
    v2f a;
    a[0] = a_base[k];
    a[1] = a_base[k + 1];
    v2f b;
    b[0] = w_base[(size_t)k * 128];
    b[1] = w_base[(size_t)(k + 1) * 128];
    c = __builtin_amdgcn_wmma_f32_16x16x4_f32(false, a, false, b, (short)0, c,
                                              false, false);
  }

  if (row0 + 16 <= n) {  // uniform fast path: no per-lane exec juggling
    float* out = H + (size_t)(row0 + (hi << 3)) * 128 + col;
#pragma unroll
    for (int r = 0; r < 8; ++r) out[(size_t)r * 128] = c[r];
  } else {
#pragma unroll
    for (int r = 0; r < 8; ++r) {
      int rr = row0 + (hi << 3) + r;
      if (rr < n) H[(size_t)rr * 128 + col] = c[r];
    }
  }
}

// ---------------- SpMM self-loop init: Y = dinv^2 * X ----------------
__global__ void spmm_selfloop_kernel(const float* __restrict__ dinv,
                                     const float* __restrict__ X,
                                     float* __restrict__ Y, int n, int F) {
  long long t = (long long)blockIdx.x * blockDim.x + threadIdx.x;
  long long total = (long long)n * F;
  if (t >= total) return;
  int i = (int)(t / F);
  float d = dinv[i];
  Y[t] = d * d * X[t];
}

// ---------------- SpMM edge scatter, F = 128: wave per edge, float4/lane ----
#define PF_DIST 1024
__global__ void spmm_edges128_kernel(const long long* __restrict__ row,
                                     const long long* __restrict__ col,
                                     const float* __restrict__ dinv,
                                     const float* __restrict__ X,
                                     float* __restrict__ Y, int e) {
  long long t = (long long)blockIdx.x * blockDim.x + threadIdx.x;
  int eid = (int)(t >> 5);
  int lane = (int)(t & 31);
  if (eid >= e) return;
  // prefetch a future edge's gather row into GL2 (random-access stream)
  int look = eid + PF_DIST;
  if (look < e) {
    int pc = (int)col[look];
    __builtin_prefetch(X + (size_t)pc * 128 + lane * 4, 0, 0);
  }
  int r = (int)row[eid], cc = (int)col[eid];
  float w = dinv[r] * dinv[cc];
  float4 v = ((const float4*)(X + (size_t)cc * 128))[lane];
  float* y = Y + (size_t)r * 128 + lane * 4;
  atomicAdd(y + 0, w * v.x);
  atomicAdd(y + 1, w * v.y);
  atomicAdd(y + 2, w * v.z);
  atomicAdd(y + 3, w * v.w);
}

// ---------------- SpMM edge scatter, F = 64: wave per edge, float2/lane ----
__global__ void spmm_edges64_kernel(const long long* __restrict__ row,
                                    const long long* __restrict__ col,
                                    const float* __restrict__ dinv,
                                    const float* __restrict__ X,
                                    float* __restrict__ Y, int e) {
  long long t = (long long)blockIdx.x * blockDim.x + threadIdx.x;
  int eid = (int)(t >> 5);
  int lane = (int)(t & 31);
  if (eid >= e) return;
  int look = eid + PF_DIST;
  if (look < e) {
    int pc = (int)col[look];
    __builtin_prefetch(X + (size_t)pc * 64 + lane * 2, 0, 0);
  }
  int r = (int)row[eid], cc = (int)col[eid];
  float w = dinv[r] * dinv[cc];
  float2 v = ((const float2*)(X + (size_t)cc * 64))[lane];
  float* y = Y + (size_t)r * 64 + lane * 2;
  atomicAdd(y + 0, w * v.x);
  atomicAdd(y + 1, w * v.y);
}

// ---------------- row-wise L2 normalize (in place), wave per row ----------
__global__ void l2norm_kernel(float* __restrict__ H, int n) {
  int t = blockIdx.x * blockDim.x + threadIdx.x;
  int rowi = t >> 5, lane = t & 31;
  if (rowi >= n) return;
  float4* p = (float4*)(H + (size_t)rowi * 128);
  float4 v = p[lane];
  float s = v.x * v.x + v.y * v.y + v.z * v.z + v.w * v.w;
#pragma unroll
  for (int off = 16; off > 0; off >>= 1) s += __shfl_down(s, off, 32);
  s = __shfl(s, 0, 32);
  float sc = 1.0f / fmaxf(sqrtf(s), 1e-12f);
  v.x *= sc; v.y *= sc; v.z *= sc; v.w *= sc;
  p[lane] = v;
}

// ---------------- GEMM2: out = [h | noise] @ W_last + b  (N x 192 x 40) ----
// block = 96 threads = 3 waves; waves handle column tiles 0..2 (40 cols
// padded to 48; pad columns read zeros via cndmask so EXEC stays all-ones).
__global__ void __launch_bounds__(96)
gemm_out_wmma(const float* __restrict__ Hn, const float* __restrict__ Nz,
              const float* __restrict__ W, const float* __restrict__ bias,
              float* __restrict__ Out, int n) {
  __shared__ float As[16 * 192];  // 12 KB
  int row0 = blockIdx.x << 4;
  if (row0 >= n) return;

  for (int t = threadIdx.x; t < 16 * 192; t += 96) {
    int r = t / 192, cidx = t - (t / 192) * 192;
    int rr = min(row0 + r, n - 1);
    As[t] = (cidx < 128) ? Hn[(size_t)rr * 128 + cidx]
                         : Nz[(size_t)rr * 64 + (cidx - 128)];
  }
  __syncthreads();

  int lane = threadIdx.x & 31;
  int wave = threadIdx.x >> 5;  // col tile 0..2
  int lm = lane & 15, hi = lane >> 4;
  int col = (wave << 4) + lm;
  bool valid = col < 40;
  int cclamp = valid ? col : 39;  // clamped address; result selected out

  v8f c;
  float b0 = valid ? bias[cclamp] : 0.0f;
#pragma unroll
  for (int r = 0; r < 8; ++r) c[r] = b0;

  const float* a_base = As + lm * 192 + (hi << 1);
  for (int k = 0; k < 192; k += 4) {
    v2f a;
    a[0] = a_base[k];
    a[1] = a_base[k + 1];
    int kk = k + (hi << 1);
    float t0 = W[kk * 40 + cclamp];
    float t1 = W[(kk + 1) * 40 + cclamp];
    v2f b;
    b[0] = valid ? t0 : 0.0f;
    b[1] = valid ? t1 : 0.0f;
    c = __builtin_amdgcn_wmma_f32_16x16x4_f32(false, a, false, b, (short)0, c,
                                              false, false);
  }

  if (valid) {
    if (row0 + 16 <= n) {  // uniform fast path
      float* out = Out + (size_t)(row0 + (hi << 3)) * 40 + col;
#pragma unroll
      for (int r = 0; r < 8; ++r) out[(size_t)r * 40] = c[r];
    } else {
#pragma unroll
      for (int r = 0; r < 8; ++r) {
        int rr = row0 + (hi << 3) + r;
        if (rr < n) Out[(size_t)rr * 40 + col] = c[r];
      }
    }
  }
}

extern "C" void kernel_launch(void* const* d_in, const int* in_sizes, int n_in,
                              void* d_out, int out_size, void* d_ws, size_t ws_size,
                              hipStream_t stream) {
  const float* x     = (const float*)d_in[0];
  const float* sf    = (const float*)d_in[1];
  const float* Wsgc  = (const float*)d_in[2];
  const float* bsgc  = (const float*)d_in[3];
  const float* Wlast = (const float*)d_in[4];
  const float* blast = (const float*)d_in[5];
  const long long* ei = (const long long*)d_in[6];  // int64 per reference

  const int n = in_sizes[1] / 64;   // N from stochastic_feature (N,64)
  const int e = in_sizes[6] / 2;    // E from edge_index (2,E)
  const long long* erow = ei;
  const long long* ecol = ei + e;

  // workspace: dinv | bufA(128) | bufB(128) | nA(64) | nB(64)
  float* ws   = (float*)d_ws;
  float* dinv = ws;
  float* bufA = dinv + n;
  float* bufB = bufA + (size_t)n * 128;
  float* nA   = bufB + (size_t)n * 128;
  float* nB   = nA + (size_t)n * 64;

  dim3 blk(256);
  int gn = (n + 255) / 256;
  int gedge = (e + 255) / 256;

  // normalized adjacency weights
  init_deg_kernel<<<gn, blk, 0, stream>>>(dinv, n);
  edge_deg_kernel<<<gedge, blk, 0, stream>>>(erow, dinv, e);
  dinv_kernel<<<gn, blk, 0, stream>>>(dinv, n);

  // h0 = x @ W_sgc + b
  gemm_sgc_wmma<<<(n + 15) / 16, 256, 0, stream>>>(x, Wsgc, bsgc, bufA, n);

  long long tot128 = (long long)n * 128;
  long long tot64  = (long long)n * 64;
  int g128 = (int)((tot128 + 255) / 256);
  int g64  = (int)((tot64 + 255) / 256);
  long long eth = (long long)e * 32;
  int ge = (int)((eth + 255) / 256);

  // h: two propagation layers (ping-pong A<->B)
  spmm_selfloop_kernel<<<g128, blk, 0, stream>>>(dinv, bufA, bufB, n, 128);
  spmm_edges128_kernel<<<ge, blk, 0, stream>>>(erow, ecol, dinv, bufA, bufB, e);
  spmm_selfloop_kernel<<<g128, blk, 0, stream>>>(dinv, bufB, bufA, n, 128);
  spmm_edges128_kernel<<<ge, blk, 0, stream>>>(erow, ecol, dinv, bufB, bufA, e);

  // noise: two propagation layers
  spmm_selfloop_kernel<<<g64, blk, 0, stream>>>(dinv, sf, nA, n, 64);
  spmm_edges64_kernel<<<ge, blk, 0, stream>>>(erow, ecol, dinv, sf, nA, e);
  spmm_selfloop_kernel<<<g64, blk, 0, stream>>>(dinv, nA, nB, n, 64);
  spmm_edges64_kernel<<<ge, blk, 0, stream>>>(erow, ecol, dinv, nA, nB, e);

  // L2 normalize h (in place)
  l2norm_kernel<<<(n * 32 + 255) / 256, blk, 0, stream>>>(bufA, n);

  // out = [h | noise] @ W_last + b_last
  gemm_out_wmma<<<(n + 15) / 16, 96, 0, stream>>>(bufA, nB, Wlast, blast,
                                                  (float*)d_out, n);
}